// HolonomicTransformerLayer_55602646614318
// MI455X (gfx1250) — compile-verified
//
#include <hip/hip_runtime.h>
#include <cstdint>

// Field-aware normalization, MI455X (gfx1250, wave32).
// One wave per 1024-float row; embeddings staged global->LDS via the CDNA5
// async data-mover path (global_load_async_to_lds_b128 + s_wait_asynccnt);
// curvature streamed with non-temporal b128 loads; output written with
// non-temporal b128 stores. Bandwidth-bound: ~403 MB total traffic ->
// ~17 us floor at 23.3 TB/s.

typedef float v4f __attribute__((ext_vector_type(4)));

#define FEAT 1024
#define WAVES_PER_BLOCK 8
#define LANES 32
#define CHUNKS 8   // 32 lanes * 8 chunks * 4 floats = 1024
#define EPS 1e-6f

// The async-to-LDS instruction adds IOFFSET to BOTH the per-lane global
// address and the per-lane LDS address, and our chunk stride is 512 B on
// both sides -> a single (ldsBase, gBase) pair serves all 8 DMAs.
#define ASYNC_LD_B128(LDS, GA, OFF)                                         \
    asm volatile("global_load_async_to_lds_b128 %0, %1, off offset:" #OFF   \
                 :: "v"(LDS), "v"(GA) : "memory")

__global__ __launch_bounds__(256) void fieldnorm_kernel(
    const float* __restrict__ emb,
    const float* __restrict__ curv,
    const float* __restrict__ gamma,
    const float* __restrict__ beta,
    const float* __restrict__ cscale,
    float* __restrict__ out,
    int nrows)
{
    // 4 KB per wave of embedding staging; 32 KB per 256-thread block
    // (well under 320 KB/WGP -> plenty of resident waves for latency hiding).
    __shared__ __align__(16) float eBuf[WAVES_PER_BLOCK * FEAT];

    const int lane = threadIdx.x & (LANES - 1);
    const int wave = threadIdx.x >> 5;
    const int row  = blockIdx.x * WAVES_PER_BLOCK + wave;
    if (row >= nrows) return;   // uniform per-wave, no intra-wave divergence

    const float* eRow = emb  + (size_t)row * FEAT;
    const float* cRow = curv + (size_t)row * FEAT;

    // ---- Stage embeddings row into LDS via async DMA. Each lane owns 16 B
    // per chunk; a full wave instruction moves 512 B coalesced.
    {
        const uint32_t ldsBase =
            (uint32_t)(uintptr_t)&eBuf[wave * FEAT + lane * 4]; // low 32b = LDS offset
        const uint64_t gBase = (uint64_t)(uintptr_t)(eRow + lane * 4);
        ASYNC_LD_B128(ldsBase, gBase, 0);
        ASYNC_LD_B128(ldsBase, gBase, 512);
        ASYNC_LD_B128(ldsBase, gBase, 1024);
        ASYNC_LD_B128(ldsBase, gBase, 1536);
        ASYNC_LD_B128(ldsBase, gBase, 2048);
        ASYNC_LD_B128(ldsBase, gBase, 2560);
        ASYNC_LD_B128(ldsBase, gBase, 3072);
        ASYNC_LD_B128(ldsBase, gBase, 3584);
    }

    // ---- While the DMAs are in flight, stream curvature directly
    // (non-temporal: touched once, 403 MB working set >> 192 MB L2)
    // and accumulate its sum of squares.
    float csq = 0.f;
#pragma unroll
    for (int j = 0; j < CHUNKS; ++j) {
        const int idx = (j * LANES + lane) * 4;
        v4f c = __builtin_nontemporal_load((const v4f*)(cRow + idx));
        csq += c.x * c.x + c.y * c.y + c.z * c.z + c.w * c.w;
    }

    // ---- Drain our async transfers; this wave only reads its own LDS
    // buffer, so no workgroup barrier is required.
    asm volatile("s_wait_asynccnt 0x0" ::: "memory");

    // ---- First pass over staged embeddings: accumulate sum / sum of
    // squares, keep values resident in VGPRs for the apply pass.
    v4f ev[CHUNKS];
    float sum = 0.f, sumsq = 0.f;
#pragma unroll
    for (int j = 0; j < CHUNKS; ++j) {
        const int idx = (j * LANES + lane) * 4;
        v4f e = *(const v4f*)&eBuf[wave * FEAT + idx];
        ev[j] = e;
        sum   += e.x + e.y + e.z + e.w;
        sumsq += e.x * e.x + e.y * e.y + e.z * e.z + e.w * e.w;
    }

    // ---- wave32 butterfly reduction (5 steps) for the three row statistics.
#pragma unroll
    for (int m = 16; m >= 1; m >>= 1) {
        sum   += __shfl_xor(sum,   m, LANES);
        sumsq += __shfl_xor(sumsq, m, LANES);
        csq   += __shfl_xor(csq,   m, LANES);
    }

    const float mean   = sum * (1.0f / FEAT);
    float var          = sumsq * (1.0f / FEAT) - mean * mean;
    var                = fmaxf(var, 0.0f);
    const float invstd = rsqrtf(var + EPS);
    const float cmag   = sqrtf(csq + EPS);
    const float factor = 1.0f / (1.0f + cscale[0] * cmag);
    const float s      = invstd * factor;   // fused scale

    // ---- Apply: out = (e - mean) * s * gamma + beta, non-temporal b128 stores.
    float* oRow = out + (size_t)row * FEAT;
#pragma unroll
    for (int j = 0; j < CHUNKS; ++j) {
        const int idx = (j * LANES + lane) * 4;
        v4f g = *(const v4f*)(gamma + idx);   // reused by every row: keep cached
        v4f b = *(const v4f*)(beta  + idx);
        v4f e = ev[j];
        v4f o;
        o.x = (e.x - mean) * s * g.x + b.x;
        o.y = (e.y - mean) * s * g.y + b.y;
        o.z = (e.z - mean) * s * g.z + b.z;
        o.w = (e.w - mean) * s * g.w + b.w;
        __builtin_nontemporal_store(o, (v4f*)(oRow + idx));
    }
}

extern "C" void kernel_launch(void* const* d_in, const int* in_sizes, int n_in,
                              void* d_out, int out_size, void* d_ws, size_t ws_size,
                              hipStream_t stream) {
    const float* emb    = (const float*)d_in[0];
    const float* curv   = (const float*)d_in[1];
    const float* gamma  = (const float*)d_in[2];
    const float* beta   = (const float*)d_in[3];
    const float* cscale = (const float*)d_in[4];
    float* out = (float*)d_out;

    const int nrows  = in_sizes[0] / FEAT;                 // B*S = 32768
    const int blocks = (nrows + WAVES_PER_BLOCK - 1) / WAVES_PER_BLOCK;

    fieldnorm_kernel<<<blocks, 256, 0, stream>>>(emb, curv, gamma, beta, cscale,
                                                 out, nrows);
}